// GraphAttentionEncoder_55035710931116
// MI455X (gfx1250) — compile-verified
//
#include <hip/hip_runtime.h>

#define N_NODES 50000
#define N_EDGES 800000
#define D_IN    256
#define HEADS   4
#define C_OUT   64
#define HC      256      // HEADS * C_OUT
#define NEG_SLOPE 0.2f

typedef __attribute__((ext_vector_type(2))) float v2f;
typedef __attribute__((ext_vector_type(8))) float v8f;

// ---------------------------------------------------------------------------
// XT[M,256] = X[M,K] @ W[K,256] + b   (fp32 WMMA)
// One block = one 16-row tile; each of its 4 waves owns a 16x64 strip and
// holds 4 accumulators, reusing one A fragment across 4 WMMAs per K-step.
// ---------------------------------------------------------------------------
__global__ void gemm_wmma_f32(const float* __restrict__ X,
                              const float* __restrict__ W,
                              const float* __restrict__ bvec,
                              float* __restrict__ XT,
                              int M, int K) {
  const int lane  = threadIdx.x & 31;
  const int wave  = threadIdx.x >> 5;          // 0..3 -> 64-col strip
  const int mt    = blockIdx.x;                // 16-row tile
  const int ncol0 = wave * 64;
  const int nlane = lane & 15;
  const int row   = mt * 16 + nlane;           // A rows: lanes 0-15 == 16-31
  const int koff  = (lane >> 4) << 1;          // lanes>=16 hold K=2,3 slice

  const float* xrow = X + (size_t)row * K;
  v8f acc0 = {}, acc1 = {}, acc2 = {}, acc3 = {};

  for (int k0 = 0; k0 < K; k0 += 4) {
    v2f a;
    a.x = xrow[k0 + koff];
    a.y = xrow[k0 + koff + 1];
    const float* w0 = W + (size_t)(k0 + koff)     * HC + ncol0 + nlane;
    const float* w1 = W + (size_t)(k0 + koff + 1) * HC + ncol0 + nlane;
    v2f b0, b1, b2, b3;
    b0.x = w0[0];  b0.y = w1[0];
    b1.x = w0[16]; b1.y = w1[16];
    b2.x = w0[32]; b2.y = w1[32];
    b3.x = w0[48]; b3.y = w1[48];
    acc0 = __builtin_amdgcn_wmma_f32_16x16x4_f32(false, a, false, b0, (short)0, acc0, false, false);
    acc1 = __builtin_amdgcn_wmma_f32_16x16x4_f32(false, a, false, b1, (short)0, acc1, false, false);
    acc2 = __builtin_amdgcn_wmma_f32_16x16x4_f32(false, a, false, b2, (short)0, acc2, false, false);
    acc3 = __builtin_amdgcn_wmma_f32_16x16x4_f32(false, a, false, b3, (short)0, acc3, false, false);
  }

  // C/D layout: VGPR v -> row = v + 8*(lane>>4), col = lane&15
  const int   rbase = mt * 16 + ((lane >> 4) << 3);
  const float bb0 = bvec[ncol0 + nlane];
  const float bb1 = bvec[ncol0 + nlane + 16];
  const float bb2 = bvec[ncol0 + nlane + 32];
  const float bb3 = bvec[ncol0 + nlane + 48];
#pragma unroll
  for (int v = 0; v < 8; ++v) {
    float* orow = XT + (size_t)(rbase + v) * HC + ncol0 + nlane;
    orow[0]  = acc0[v] + bb0;
    orow[16] = acc1[v] + bb1;
    orow[32] = acc2[v] + bb2;
    orow[48] = acc3[v] + bb3;
  }
}

// ---------------------------------------------------------------------------
// float atomic max via sign-aware integer atomics (valid with -inf init)
// ---------------------------------------------------------------------------
__device__ inline void atomicMaxFloat(float* addr, float val) {
  if (val >= 0.f) atomicMax((int*)addr, __float_as_int(val));
  else            atomicMin((unsigned int*)addr, __float_as_uint(val));
}

// ---------------------------------------------------------------------------
// init: nmax = -inf, denom = 0, accum = 0
// ---------------------------------------------------------------------------
__global__ void init_buffers(float* __restrict__ nmax,
                             float* __restrict__ denom,
                             float* __restrict__ accum) {
  int idx = blockIdx.x * blockDim.x + threadIdx.x;
  if (idx < N_NODES * HEADS) { nmax[idx] = -__builtin_inff(); denom[idx] = 0.f; }
  if (idx < N_NODES * HC)    accum[idx] = 0.f;
}

// ---------------------------------------------------------------------------
// pass 1: per-edge logits + segment max.  One wave per edge: lane l handles
// channels [8l, 8l+8) -> all inside head l/8 (64 ch/head).  float4 loads.
// ---------------------------------------------------------------------------
__global__ void edge_logits(const float* __restrict__ XT,
                            const int* __restrict__ src,
                            const int* __restrict__ dst,
                            const float* __restrict__ att,   // [H*C] = 256
                            float* __restrict__ logits,      // [E, H]
                            float* __restrict__ nmax) {      // [N, H]
  const int lane = threadIdx.x & 31;
  const int e = blockIdx.x * (blockDim.x >> 5) + (threadIdx.x >> 5);
  if (e >= N_EDGES) return;
  const int s = src[e], d = dst[e];
  const int cb = lane * 8;                                   // 32B aligned
  const float4* xi4 = (const float4*)(XT + (size_t)d * HC + cb);
  const float4* xj4 = (const float4*)(XT + (size_t)s * HC + cb);
  const float4* at4 = (const float4*)(att + cb);
  float p = 0.f;
#pragma unroll
  for (int q = 0; q < 2; ++q) {
    float4 xi = xi4[q], xj = xj4[q], at = at4[q];
    float v;
    v = xi.x + xj.x; v = v > 0.f ? v : NEG_SLOPE * v; p += v * at.x;
    v = xi.y + xj.y; v = v > 0.f ? v : NEG_SLOPE * v; p += v * at.y;
    v = xi.z + xj.z; v = v > 0.f ? v : NEG_SLOPE * v; p += v * at.z;
    v = xi.w + xj.w; v = v > 0.f ? v : NEG_SLOPE * v; p += v * at.w;
  }
  // reduce within 8-lane groups (one head per group)
  p += __shfl_xor(p, 1, 32);
  p += __shfl_xor(p, 2, 32);
  p += __shfl_xor(p, 4, 32);
  if ((lane & 7) == 0) {
    const int h = lane >> 3;
    logits[(size_t)e * HEADS + h] = p;
    atomicMaxFloat(&nmax[(size_t)d * HEADS + h], p);
  }
}

// ---------------------------------------------------------------------------
// pass 2: ex = exp(logit - max[dst]);  denom[dst] += ex  (one thread per e*h)
// ---------------------------------------------------------------------------
__global__ void edge_exp(const int* __restrict__ dst,
                         const float* __restrict__ nmax,
                         float* __restrict__ logits,          // in-place -> ex
                         float* __restrict__ denom) {
  int idx = blockIdx.x * blockDim.x + threadIdx.x;
  if (idx >= N_EDGES * HEADS) return;
  int e = idx >> 2, h = idx & 3;
  int d = dst[e];
  float ex = __expf(logits[idx] - nmax[d * HEADS + h]);
  logits[idx] = ex;
  atomicAdd(&denom[d * HEADS + h], ex);
}

// ---------------------------------------------------------------------------
// pass 3: accum[dst] += xj * alpha   (one wave per edge, 8 ch/lane)
// ---------------------------------------------------------------------------
__global__ void edge_scatter(const float* __restrict__ XT,
                             const int* __restrict__ src,
                             const int* __restrict__ dst,
                             const float* __restrict__ ex,     // [E,H]
                             const float* __restrict__ denom,  // [N,H]
                             float* __restrict__ accum) {      // [N,HC]
  const int lane = threadIdx.x & 31;
  const int e = blockIdx.x * (blockDim.x >> 5) + (threadIdx.x >> 5);
  if (e >= N_EDGES) return;
  const int s = src[e], d = dst[e];
  const int h = lane >> 3;
  const float alpha = ex[(size_t)e * HEADS + h] /
                      (denom[(size_t)d * HEADS + h] + 1e-16f);
  const float4* xj4 = (const float4*)(XT + (size_t)s * HC + lane * 8);
  float*        out = accum + (size_t)d * HC + lane * 8;
  float4 a = xj4[0], b = xj4[1];
  atomicAdd(&out[0], a.x * alpha);
  atomicAdd(&out[1], a.y * alpha);
  atomicAdd(&out[2], a.z * alpha);
  atomicAdd(&out[3], a.w * alpha);
  atomicAdd(&out[4], b.x * alpha);
  atomicAdd(&out[5], b.y * alpha);
  atomicAdd(&out[6], b.z * alpha);
  atomicAdd(&out[7], b.w * alpha);
}

// ---------------------------------------------------------------------------
// finalize: head mean + bias + relu -> [N, C_OUT]
// ---------------------------------------------------------------------------
__global__ void head_mean(const float* __restrict__ accum,
                          const float* __restrict__ bias,
                          float* __restrict__ out) {
  int idx = blockIdx.x * blockDim.x + threadIdx.x;
  if (idx >= N_NODES * C_OUT) return;
  int n = idx / C_OUT, c = idx - n * C_OUT;
  const float* a = accum + (size_t)n * HC + c;
  float v = 0.25f * (a[0] + a[64] + a[128] + a[192]) + bias[c];
  out[idx] = v > 0.f ? v : 0.f;
}

// ---------------------------------------------------------------------------
extern "C" void kernel_launch(void* const* d_in, const int* in_sizes, int n_in,
                              void* d_out, int out_size, void* d_ws, size_t ws_size,
                              hipStream_t stream) {
  const float* x     = (const float*)d_in[0];
  const int*   ei    = (const int*)  d_in[1];
  const float* W1    = (const float*)d_in[2];
  const float* b1    = (const float*)d_in[3];
  const float* att1  = (const float*)d_in[4];
  const float* bias1 = (const float*)d_in[5];
  const float* W2    = (const float*)d_in[6];
  const float* b2    = (const float*)d_in[7];
  const float* att2  = (const float*)d_in[8];
  const float* bias2 = (const float*)d_in[9];
  const int* src = ei;             // edge_index[0]
  const int* dst = ei + N_EDGES;   // edge_index[1]

  float* ws = (float*)d_ws;
  float* XT    = ws;  ws += (size_t)N_NODES * HC;      // 51.2 MB (L2-resident)
  float* accum = ws;  ws += (size_t)N_NODES * HC;      // 51.2 MB
  float* logit = ws;  ws += (size_t)N_EDGES * HEADS;   // 12.8 MB
  float* nmax  = ws;  ws += (size_t)N_NODES * HEADS;
  float* denom = ws;  ws += (size_t)N_NODES * HEADS;
  float* h1    = ws;  ws += (size_t)N_NODES * C_OUT;

  const int mtiles  = N_NODES / 16;                    // 3125 (exact)
  const int edgeBlk = (N_EDGES + 7) / 8;               // 1 wave per edge
  const int initN   = N_NODES * HC;

  // ----- layer 1 -----
  gemm_wmma_f32<<<mtiles, 128, 0, stream>>>(x, W1, b1, XT, N_NODES, D_IN);
  init_buffers <<<(initN + 255) / 256, 256, 0, stream>>>(nmax, denom, accum);
  edge_logits  <<<edgeBlk, 256, 0, stream>>>(XT, src, dst, att1, logit, nmax);
  edge_exp     <<<(N_EDGES * HEADS + 255) / 256, 256, 0, stream>>>(dst, nmax, logit, denom);
  edge_scatter <<<edgeBlk, 256, 0, stream>>>(XT, src, dst, logit, denom, accum);
  head_mean    <<<(N_NODES * C_OUT + 255) / 256, 256, 0, stream>>>(accum, bias1, h1);

  // ----- layer 2 -----
  gemm_wmma_f32<<<mtiles, 128, 0, stream>>>(h1, W2, b2, XT, N_NODES, C_OUT);
  init_buffers <<<(initN + 255) / 256, 256, 0, stream>>>(nmax, denom, accum);
  edge_logits  <<<edgeBlk, 256, 0, stream>>>(XT, src, dst, att2, logit, nmax);
  edge_exp     <<<(N_EDGES * HEADS + 255) / 256, 256, 0, stream>>>(dst, nmax, logit, denom);
  edge_scatter <<<edgeBlk, 256, 0, stream>>>(XT, src, dst, logit, denom, accum);
  head_mean    <<<(N_NODES * C_OUT + 255) / 256, 256, 0, stream>>>(accum, bias2, (float*)d_out);
}